// DGLGCN_87952340287676
// MI455X (gfx1250) — compile-verified
//
#include <hip/hip_runtime.h>

// ---------------------------------------------------------------------------
// DGL-style 2-layer GCN + dot-product scoring for MI455X (gfx1250, wave32).
//
//   layer: h = tanh( mean_aggregate(h, src->dst) @ W^T + b )   (twice)
//   out[b,k] = <h[user_index[b,k]], h[item_index[b,k]]>
//
// Bottleneck is the edge scatter (~1.3 GB of L2/HBM traffic); the 64x64
// linear layers run through V_WMMA_F32_16X16X4_F32 (f32-exact vs reference).
// ---------------------------------------------------------------------------

#define N_NODES 50000
#define EMBED   64
#define N_EDGES 1250000
#define N_SCORE (1024 * 100)
#define N_TILES (N_NODES / 16)        // 3125 row tiles of 16 (divides exactly)

typedef __attribute__((ext_vector_type(2))) float v2f;
typedef __attribute__((ext_vector_type(8))) float v8f;

// --------------------------- workspace zero fill ---------------------------
__global__ void gcn_zero_kernel(float4* __restrict__ p, long n4) {
    long i = (long)blockIdx.x * blockDim.x + threadIdx.x;
    if (i < n4) p[i] = make_float4(0.f, 0.f, 0.f, 0.f);
}

// ------------------------------ degree count -------------------------------
__global__ void gcn_degree_kernel(const int* __restrict__ dst,
                                  float* __restrict__ deg) {
    int i = blockIdx.x * blockDim.x + threadIdx.x;
    if (i < N_EDGES) unsafeAtomicAdd(&deg[dst[i]], 1.0f);   // global_atomic_add_f32
}

// ------------------------ edge gather + scatter-add ------------------------
// 16 lanes per edge, each lane moves a float4 of the 64-float row.
__global__ void gcn_scatter_kernel(const float* __restrict__ h,
                                   const int* __restrict__ src,
                                   const int* __restrict__ dst,
                                   float* __restrict__ msg) {
    long gid = (long)blockIdx.x * blockDim.x + threadIdx.x;
    int  e   = (int)(gid >> 4);
    if (e >= N_EDGES) return;
    int  c   = ((int)gid & 15) * 4;
    const float4 v = *(const float4*)(h + (size_t)src[e] * EMBED + c);
    float* d = msg + (size_t)dst[e] * EMBED + c;
    unsafeAtomicAdd(d + 0, v.x);
    unsafeAtomicAdd(d + 1, v.y);
    unsafeAtomicAdd(d + 2, v.z);
    unsafeAtomicAdd(d + 3, v.w);
}

// ---------------- node update: mean / Linear(W^T,b) / tanh -----------------
// One wave32 per 16-node tile. out[n,j] = tanh( sum_k (msg[n,k]/deg[n]) * W[j,k] + b[j] )
// A-fragment (16x4 f32): lanes 0-15 -> K=k0,k0+1 ; lanes 16-31 -> K=k0+2,k0+3.
// B-fragment (4x16 f32): VGPR0 = row k0 (lanes 0-15) / row k0+2 (lanes 16-31),
//                        VGPR1 = row k0+1 / k0+3.  B = W^T so each lane reads
//                        W[(j0+ln)*64 + k0 + 2*khalf .. +1] as a float2.
// In-place update is safe: a wave reads all 16 of its rows before storing them.
__global__ __launch_bounds__(256)
void gcn_update_kernel(float* __restrict__ hbuf, const float* __restrict__ deg,
                       const float* __restrict__ W, const float* __restrict__ bias) {
    const int wave = threadIdx.x >> 5;
    const int lane = threadIdx.x & 31;
    const int tile = blockIdx.x * 8 + wave;
    if (tile >= N_TILES) return;            // wave-uniform: EXEC stays all-ones
    const int ln    = lane & 15;
    const int khalf = lane >> 4;
    const int m0    = tile * 16;

    const int   arow_idx = m0 + ln;
    const float inv  = 1.0f / fmaxf(deg[arow_idx], 1.0f);   // fused mean
    const float* arow = hbuf + (size_t)arow_idx * EMBED + khalf * 2;
    const float* wrow = W    + (size_t)ln       * EMBED + khalf * 2;

    v8f acc0 = {}, acc1 = {}, acc2 = {}, acc3 = {};
#pragma unroll
    for (int k0 = 0; k0 < EMBED; k0 += 4) {
        float2 af = *(const float2*)(arow + k0);
        v2f a;  a.x = af.x * inv;  a.y = af.y * inv;
        float2 bf0 = *(const float2*)(wrow +  0 * EMBED + k0);
        float2 bf1 = *(const float2*)(wrow + 16 * EMBED + k0);
        float2 bf2 = *(const float2*)(wrow + 32 * EMBED + k0);
        float2 bf3 = *(const float2*)(wrow + 48 * EMBED + k0);
        v2f b0v = {bf0.x, bf0.y};
        v2f b1v = {bf1.x, bf1.y};
        v2f b2v = {bf2.x, bf2.y};
        v2f b3v = {bf3.x, bf3.y};
        acc0 = __builtin_amdgcn_wmma_f32_16x16x4_f32(false, a, false, b0v, (short)0, acc0, false, false);
        acc1 = __builtin_amdgcn_wmma_f32_16x16x4_f32(false, a, false, b1v, (short)0, acc1, false, false);
        acc2 = __builtin_amdgcn_wmma_f32_16x16x4_f32(false, a, false, b2v, (short)0, acc2, false, false);
        acc3 = __builtin_amdgcn_wmma_f32_16x16x4_f32(false, a, false, b3v, (short)0, acc3, false, false);
    }

    // C/D 16x16 f32 layout: VGPR r -> row (m0 + khalf*8 + r), col = ln per tile.
#pragma unroll
    for (int r = 0; r < 8; ++r) {
        float* orow = hbuf + (size_t)(m0 + khalf * 8 + r) * EMBED;
        orow[ 0 + ln] = tanhf(acc0[r] + bias[ 0 + ln]);
        orow[16 + ln] = tanhf(acc1[r] + bias[16 + ln]);
        orow[32 + ln] = tanhf(acc2[r] + bias[32 + ln]);
        orow[48 + ln] = tanhf(acc3[r] + bias[48 + ln]);
    }
}

// ----------------------------- scoring kernel ------------------------------
__global__ void gcn_score_kernel(const float* __restrict__ h,
                                 const int* __restrict__ ui,
                                 const int* __restrict__ ii,
                                 float* __restrict__ out) {
    int i = blockIdx.x * blockDim.x + threadIdx.x;
    if (i >= N_SCORE) return;
    const float4* u = (const float4*)(h + (size_t)ui[i] * EMBED);
    const float4* v = (const float4*)(h + (size_t)ii[i] * EMBED);
    float s = 0.f;
#pragma unroll
    for (int c = 0; c < 16; ++c) {
        float4 a = u[c], b = v[c];
        s += a.x * b.x + a.y * b.y + a.z * b.z + a.w * b.w;
    }
    out[i] = s;
}

// ------------------------------- launcher ----------------------------------
extern "C" void kernel_launch(void* const* d_in, const int* in_sizes, int n_in,
                              void* d_out, int out_size, void* d_ws, size_t ws_size,
                              hipStream_t stream) {
    const float* emb = (const float*)d_in[0];
    const float* W0  = (const float*)d_in[1];
    const float* b0  = (const float*)d_in[2];
    const float* W1  = (const float*)d_in[3];
    const float* b1  = (const float*)d_in[4];
    const int*   src = (const int*)d_in[5];
    const int*   dst = (const int*)d_in[6];
    const int*   ui  = (const int*)d_in[7];
    const int*   ii  = (const int*)d_in[8];
    float* out = (float*)d_out;

    // Workspace layout (floats): deg[50000 pad->51200] | msgA[50000*64] | msgB[50000*64]
    float* deg  = (float*)d_ws;
    float* msgA = deg  + 51200;
    float* msgB = msgA + (size_t)N_NODES * EMBED;
    const long totalFloats = 51200 + 2L * N_NODES * EMBED;   // 6,451,200

    gcn_zero_kernel<<<(int)((totalFloats / 4 + 255) / 256), 256, 0, stream>>>(
        (float4*)d_ws, totalFloats / 4);

    gcn_degree_kernel<<<(N_EDGES + 255) / 256, 256, 0, stream>>>(dst, deg);

    const int scatterBlocks = (int)(((long)N_EDGES * 16 + 255) / 256);  // 78125
    const int updateBlocks  = (N_TILES + 7) / 8;                        // 391

    // Layer 0: emb --scatter--> msgA --(mean,W0,b0,tanh in-place)--> h1 = msgA
    gcn_scatter_kernel<<<scatterBlocks, 256, 0, stream>>>(emb, src, dst, msgA);
    gcn_update_kernel<<<updateBlocks, 256, 0, stream>>>(msgA, deg, W0, b0);

    // Layer 1: h1 --scatter--> msgB --(mean,W1,b1,tanh in-place)--> h2 = msgB
    gcn_scatter_kernel<<<scatterBlocks, 256, 0, stream>>>(msgA, src, dst, msgB);
    gcn_update_kernel<<<updateBlocks, 256, 0, stream>>>(msgB, deg, W1, b1);

    gcn_score_kernel<<<(N_SCORE + 255) / 256, 256, 0, stream>>>(msgB, ui, ii, out);
}